// TemporalAttention_46763603919124
// MI455X (gfx1250) — compile-verified
//
#include <hip/hip_runtime.h>
#include <math.h>

// ---------------------------------------------------------------------------
// TemporalAttention, algebraically collapsed:
//   lam[b,t] = z[b,t,:] . G . z[b,T-1,:]   with  G = W W^T  (256x256, symmetric)
//   out[b,:] = softmax_t(lam[b,:]) @ z[b,:,:]
// Dominant cost: streaming z twice (2 x 268 MB) -> ~23 us at 23.3 TB/s.
// G computed once in full f32 with V_WMMA_F32_16X16X4_F32.
// All streaming loops use b128 (float4) accesses to saturate HBM.
// ---------------------------------------------------------------------------

#define B_DIM 512
#define T_DIM 512
#define D_DIM 256

typedef __attribute__((ext_vector_type(2))) float v2f;
typedef __attribute__((ext_vector_type(4))) float v4f;
typedef __attribute__((ext_vector_type(8))) float v8f;

// ---------------------------------------------------------------------------
// Kernel 1: G = W * W^T  (256x256x256, f32 WMMA).
// One wave per 16x16 output tile; 64 x V_WMMA_F32_16X16X4_F32 K-loop.
// f32 A 16x4 layout (ISA 7.12.2): lanes 0-15 -> M=lane, {K=k0,k0+1};
// lanes 16-31 -> M=lane-16, {K=k0+2,k0+3}. B 4x16 mirrors with N=lane&15.
// ---------------------------------------------------------------------------
__global__ __launch_bounds__(32) void
gram_wmma_f32_kernel(const float* __restrict__ W, float* __restrict__ G) {
    const int lane = threadIdx.x;            // 0..31, wave32
    const int tile = blockIdx.x;             // 0..255
    const int e0 = (tile >> 4) << 4;         // output row tile base
    const int f0 = (tile & 15) << 4;         // output col tile base

    const int mn = lane & 15;                // M for A, N for B
    const int kh = (lane >> 4) << 1;         // 0 or 2: K sub-pair per half-wave

    v8f c = {};
    for (int k0 = 0; k0 < D_DIM; k0 += 4) {
        v2f a = *(const v2f*)&W[(e0 + mn) * D_DIM + k0 + kh]; // A[m][k]
        v2f b = *(const v2f*)&W[(f0 + mn) * D_DIM + k0 + kh]; // B[k][n] = W[f0+n][k0+k]
        c = __builtin_amdgcn_wmma_f32_16x16x4_f32(
                false, a, false, b, (short)0, c, false, false);
    }

    // C/D 16x16 f32: VGPR v, lanes 0-15 -> M=v, lanes 16-31 -> M=v+8.
    const int row_base = e0 + ((lane >> 4) << 3);
    const int col = f0 + (lane & 15);
#pragma unroll
    for (int v = 0; v < 8; ++v)
        G[(row_base + v) * D_DIM + col] = c[v];
}

__device__ __forceinline__ float dot4(v4f a, v4f b) {
    float s = a.x * b.x;
    s = fmaf(a.y, b.y, s);
    s = fmaf(a.z, b.z, s);
    s = fmaf(a.w, b.w, s);
    return s;
}

// ---------------------------------------------------------------------------
// Kernel 2: per-batch attention. One 256-thread block (8 wave32) per batch.
// ---------------------------------------------------------------------------
__global__ __launch_bounds__(256) void
temporal_attn_kernel(const float* __restrict__ z, const float* __restrict__ G,
                     float* __restrict__ out) {
    __shared__ float zq[D_DIM];        // z[b, T-1, :]
    __shared__ float vsh[D_DIM];       // v = G * zq
    __shared__ float lam[T_DIM];       // logits -> exp weights
    __shared__ float red[256];         // block-reduction scratch
    __shared__ float osum[4 * D_DIM];  // output partials (4 T-groups)

    const int b    = blockIdx.x;
    const int tid  = threadIdx.x;      // 0..255
    const int lane = tid & 31;
    const int wave = tid >> 5;
    const float* __restrict__ zb = z + (size_t)b * T_DIM * D_DIM;

    // ---- load query row zq ----
    zq[tid] = zb[(size_t)(T_DIM - 1) * D_DIM + tid];
    __syncthreads();

    const v4f* __restrict__ zqv  = (const v4f*)zq;
    const v4f* __restrict__ vshv = (const v4f*)vsh;

    // ---- v[e] = G[e,:] . zq ; wave-per-row, 2 x b128 per lane per row ----
    for (int e = wave; e < D_DIM; e += 8) {
        const v4f* __restrict__ grow = (const v4f*)(G + (size_t)e * D_DIM);
        float s = dot4(grow[lane], zqv[lane]) + dot4(grow[lane + 32], zqv[lane + 32]);
#pragma unroll
        for (int off = 16; off > 0; off >>= 1)
            s += __shfl_xor(s, off, 32);
        if (lane == 0) vsh[e] = s;
    }
    __syncthreads();

    // ---- lam[t] = z[b,t,:] . v ; wave-per-row, 2 x b128 per lane per row ----
    for (int t = wave; t < T_DIM; t += 8) {
        const v4f* __restrict__ row = (const v4f*)(zb + (size_t)t * D_DIM);
        float s = dot4(row[lane], vshv[lane]) + dot4(row[lane + 32], vshv[lane + 32]);
#pragma unroll
        for (int off = 16; off > 0; off >>= 1)
            s += __shfl_xor(s, off, 32);
        if (lane == 0) lam[t] = s;
    }
    __syncthreads();

    // ---- softmax over t (512 values, 2 per thread) ----
    red[tid] = fmaxf(lam[tid], lam[tid + 256]);
    __syncthreads();
    for (int s = 128; s > 0; s >>= 1) {
        if (tid < s) red[tid] = fmaxf(red[tid], red[tid + s]);
        __syncthreads();
    }
    const float mx = red[0];
    __syncthreads();

    const float p0 = __expf(lam[tid] - mx);
    const float p1 = __expf(lam[tid + 256] - mx);
    lam[tid] = p0;
    lam[tid + 256] = p1;
    red[tid] = p0 + p1;
    __syncthreads();
    for (int s = 128; s > 0; s >>= 1) {
        if (tid < s) red[tid] += red[tid + s];
        __syncthreads();
    }
    const float inv_sum = 1.0f / red[0];
    __syncthreads();

    // ---- out[b,:] = (sum_t p[t] * z[b,t,:]) / sum ----
    // Thread owns 4 consecutive columns; 64-thread groups stride T by 4.
    // Every load is a coalesced b128 (512 B per wave per instruction);
    // non-temporal: this is the last consumer of z.
    {
        const int grp  = tid >> 6;          // 0..3
        const int col4 = (tid & 63) << 2;   // 0,4,...,252
        v4f acc = {0.f, 0.f, 0.f, 0.f};
        for (int t = grp; t < T_DIM; t += 4) {
            const v4f zv = __builtin_nontemporal_load(
                (const v4f*)(zb + (size_t)t * D_DIM + col4));
            const float p = lam[t];
            acc.x = fmaf(p, zv.x, acc.x);
            acc.y = fmaf(p, zv.y, acc.y);
            acc.z = fmaf(p, zv.z, acc.z);
            acc.w = fmaf(p, zv.w, acc.w);
        }
        *(v4f*)&osum[grp * D_DIM + col4] = acc;
    }
    __syncthreads();

    // combine the 4 group partials per column and store
    {
        const float o = ((osum[0 * D_DIM + tid] + osum[1 * D_DIM + tid]) +
                         (osum[2 * D_DIM + tid] + osum[3 * D_DIM + tid])) * inv_sum;
        out[(size_t)b * D_DIM + tid] = o;
    }
}

// ---------------------------------------------------------------------------
extern "C" void kernel_launch(void* const* d_in, const int* in_sizes, int n_in,
                              void* d_out, int out_size, void* d_ws, size_t ws_size,
                              hipStream_t stream) {
    const float* z = (const float*)d_in[0];   // (512, 512, 256) f32
    const float* W = (const float*)d_in[1];   // (256, 256) f32
    float* out = (float*)d_out;               // (512, 256) f32
    float* G   = (float*)d_ws;                // 256*256*4 = 256 KB scratch

    gram_wmma_f32_kernel<<<(D_DIM / 16) * (D_DIM / 16), 32, 0, stream>>>(W, G);
    temporal_attn_kernel<<<B_DIM, 256, 0, stream>>>(z, G, out);
}